// UniformAffineQuantizer_15796889714962
// MI455X (gfx1250) — compile-verified
//
#include <hip/hip_runtime.h>
#include <stdint.h>

// ---------------------------------------------------------------------------
// Per-group fake quantization (UniformAffineQuantizer steady-state path).
//   x:[4096,11008] f32, alpha/shift:[NGROUPS,1] f32, perturb:[NGROUPS,128] f32
//   out = fakequant(x) f32
// Memory-bound: ~543 MB moved -> ~23 us at 23.3 TB/s. No matrix math, so no
// WMMA; CDNA5 features used: wave32 shuffle reduction, b128 global/LDS ops,
// async global->LDS double-buffer pipeline tracked with ASYNCcnt, and
// non-temporal cache hints (stream >> 192 MB L2, zero reuse).
// ---------------------------------------------------------------------------

#define QMAXF    255.0f
#define CLIPMINF 1e-5f
#define CLIPMAXF 1e4f
#define GROUP    128
#define ROWS     4096
#define COLS     11008
#define NGROUPS  (ROWS * (COLS / GROUP))   // 352256
#define WAVES_PB 8
#define THREADS  (WAVES_PB * 32)
#define BUF_BYTES (WAVES_PB * GROUP * 4)   // 4096 B per buffer per array

typedef float v4f __attribute__((ext_vector_type(4)));

template <int N>
__device__ __forceinline__ void wait_async() {
#if __has_builtin(__builtin_amdgcn_s_wait_asynccnt)
  __builtin_amdgcn_s_wait_asynccnt(N);
#else
  asm volatile("s_wait_asynccnt %0" ::"i"(N) : "memory");
#endif
}

// GLOBAL_LOAD_ASYNC_TO_LDS_B128, GVS mode, non-temporal (read-once stream):
//   lds[lds_off] = mem[sgpr_base + vgpr_off32]   (16 bytes per lane)
__device__ __forceinline__ void async_b128_nt(uint32_t lds_off, uint32_t goff,
                                              const void* sbase) {
  asm volatile("global_load_async_to_lds_b128 %0, %1, %2 th:TH_LOAD_NT"
               :
               : "v"(lds_off), "v"(goff), "s"(sbase)
               : "memory");
}

__global__ __launch_bounds__(THREADS) void UniformAffineQuantizer_kernel(
    const float* __restrict__ x, const float* __restrict__ alpha,
    const float* __restrict__ shift, const float* __restrict__ perturb,
    float* __restrict__ out) {
  __shared__ __align__(16) float xb[2][WAVES_PB][GROUP];
  __shared__ __align__(16) float pb[2][WAVES_PB][GROUP];

  const int lane = threadIdx.x & 31;
  const int wv = threadIdx.x >> 5;
  const int waveGlobal = blockIdx.x * WAVES_PB + wv;
  const int totalWaves = gridDim.x * WAVES_PB;

  // LDS byte offsets for this lane's 16B slot in each buffer (low 32 bits of
  // a generic LDS pointer are the LDS byte address).
  const uint32_t ldsX0 = (uint32_t)(uintptr_t)&xb[0][wv][lane * 4];
  const uint32_t ldsP0 = (uint32_t)(uintptr_t)&pb[0][wv][lane * 4];

  auto prefetch = [&](int g, int buf) {
    const uint32_t goff = (uint32_t)g * (GROUP * 4u) + (uint32_t)lane * 16u;
    async_b128_nt(ldsX0 + (uint32_t)buf * BUF_BYTES, goff, x);
    async_b128_nt(ldsP0 + (uint32_t)buf * BUF_BYTES, goff, perturb);
  };

  auto compute = [&](int g, int buf) {
    const float4 xv = *(const float4*)&xb[buf][wv][lane * 4];
    const float4 pv = *(const float4*)&pb[buf][wv][lane * 4];

    // group min: 4 local values then 5-step wave32 xor-shuffle reduction
    float m = fminf(fminf(xv.x, xv.y), fminf(xv.z, xv.w));
#pragma unroll
    for (int off = 16; off >= 1; off >>= 1)
      m = fminf(m, __shfl_xor(m, off, 32));

    // per-group scalars (wave-uniform -> scalar loads)
    const int gu = __builtin_amdgcn_readfirstlane(g);
    const float a = alpha[gu];
    const float sh = shift[gu];

    const float scale = fminf(fmaxf(a * (1.0f / QMAXF), CLIPMINF), CLIPMAXF);
    const float zp = -(m + sh) / scale;              // one divide per group
    const float rzp = rintf(fminf(fmaxf(zp, -CLIPMAXF), CLIPMAXF));
    const float inv = 1.0f / scale;                  // one divide per group

    v4f r;
    {
      float q;
      q = rintf((xv.x + pv.x) * inv) + rzp;
      r.x = (fminf(fmaxf(q, 0.0f), QMAXF) - rzp) * scale;
      q = rintf((xv.y + pv.y) * inv) + rzp;
      r.y = (fminf(fmaxf(q, 0.0f), QMAXF) - rzp) * scale;
      q = rintf((xv.z + pv.z) * inv) + rzp;
      r.z = (fminf(fmaxf(q, 0.0f), QMAXF) - rzp) * scale;
      q = rintf((xv.w + pv.w) * inv) + rzp;
      r.w = (fminf(fmaxf(q, 0.0f), QMAXF) - rzp) * scale;
    }
    // write-once stream: non-temporal store (keeps L2 for reused data)
    __builtin_nontemporal_store(
        r, (v4f*)(out + (size_t)g * GROUP + (size_t)lane * 4));
  };

  int g = waveGlobal;
  if (g >= NGROUPS) return;

  int cur = 0;
  prefetch(g, cur);  // ASYNCcnt = 2
  while (g < NGROUPS) {
    const int gn = g + totalWaves;
    if (gn < NGROUPS) {
      prefetch(gn, cur ^ 1);  // up to 4 in flight
      wait_async<2>();        // in-order completion => current tile landed
    } else {
      wait_async<0>();        // drain for the last tile
    }
    compute(g, cur);
    g = gn;
    cur ^= 1;
  }
}

extern "C" void kernel_launch(void* const* d_in, const int* in_sizes, int n_in,
                              void* d_out, int out_size, void* d_ws,
                              size_t ws_size, hipStream_t stream) {
  (void)in_sizes; (void)n_in; (void)out_size; (void)d_ws; (void)ws_size;
  const float* x = (const float*)d_in[0];
  const float* alpha = (const float*)d_in[1];
  const float* shift = (const float*)d_in[2];
  const float* perturb = (const float*)d_in[3];
  float* out = (float*)d_out;

  // 4096 persistent blocks x 8 waves = 32768 waves grid-striding 352256 groups
  // (~10.75 tiles/wave -> async double-buffer reaches steady state).
  const int blocks = 4096;
  UniformAffineQuantizer_kernel<<<blocks, THREADS, 0, stream>>>(
      x, alpha, shift, perturb, out);
}